// GroupLasso_79869211836511
// MI455X (gfx1250) — compile-verified
//
#include <hip/hip_runtime.h>
#include <math.h>

// Group lasso proximal operator, MI455X (gfx1250).
// Memory-bound: ~670 MB HBM traffic -> ~29 us floor at 23.3 TB/s.
// Group accumulator / factor table (65536 floats = 256 KB) is L2/LDS resident.

#define NGROUPS   65536
#define GROUP_REG 0.1f
#define STEP_SIZE 0.01f
#define EPSV      1e-10f

// ---------------------------------------------------------------- pass 0: zero
__global__ void gl_zero_kernel(float* __restrict__ ws) {
    int i = blockIdx.x * blockDim.x + threadIdx.x;   // exactly NGROUPS threads
    ws[i] = 0.0f;
}

// ------------------------------------------------------- pass 1: sum of squares
// Vectorized stream (float4/int4) + no-return FP32 global atomics (L2 atomic
// units, fire-and-forget since the result is discarded) + global_prefetch_b8.
__global__ void gl_sumsq_kernel(const float4* __restrict__ c4,
                                const int4*  __restrict__ g4,
                                float* __restrict__ sumsq,
                                int n4, int n) {
    const int stride = gridDim.x * blockDim.x;
    int tid = blockIdx.x * blockDim.x + threadIdx.x;

    for (int i = tid; i < n4; i += stride) {
        float4 c = c4[i];
        int4   g = g4[i];
        // prefetch next tile of both streams (speculative, drops on fault)
        __builtin_prefetch(&c4[i + stride], 0, 3);
        __builtin_prefetch(&g4[i + stride], 0, 3);
        __hip_atomic_fetch_add(&sumsq[g.x], c.x * c.x,
                               __ATOMIC_RELAXED, __HIP_MEMORY_SCOPE_AGENT);
        __hip_atomic_fetch_add(&sumsq[g.y], c.y * c.y,
                               __ATOMIC_RELAXED, __HIP_MEMORY_SCOPE_AGENT);
        __hip_atomic_fetch_add(&sumsq[g.z], c.z * c.z,
                               __ATOMIC_RELAXED, __HIP_MEMORY_SCOPE_AGENT);
        __hip_atomic_fetch_add(&sumsq[g.w], c.w * c.w,
                               __ATOMIC_RELAXED, __HIP_MEMORY_SCOPE_AGENT);
    }
    // scalar tail (N = 2^25 so normally empty)
    const float* c1 = (const float*)c4;
    const int*   g1 = (const int*)g4;
    for (int i = n4 * 4 + tid; i < n; i += stride) {
        float c = c1[i];
        __hip_atomic_fetch_add(&sumsq[g1[i]], c * c,
                               __ATOMIC_RELAXED, __HIP_MEMORY_SCOPE_AGENT);
    }
}

// ------------------------------------------------- pass 2: per-group factor
__global__ void gl_factor_kernel(float* __restrict__ ws) {
    int g = blockIdx.x * blockDim.x + threadIdx.x;   // exactly NGROUPS threads
    float norm   = sqrtf(ws[g]);
    float shrink = fmaxf(1.0f - (GROUP_REG * STEP_SIZE) / (norm + EPSV), 0.0f);
    ws[g] = (norm > EPSV) ? shrink : 1.0f;
}

// --- LDS gather helpers: hardware ds_load_b32, issued back-to-back with a
// single s_wait_dscnt. Early-clobber outputs: DS writebacks are asynchronous,
// so a result register must not alias a later load's address operand.
__device__ __forceinline__ void lds_gather4(unsigned a0, unsigned a1,
                                            unsigned a2, unsigned a3,
                                            float& f0, float& f1,
                                            float& f2, float& f3) {
    asm volatile("ds_load_b32 %0, %4\n\t"
                 "ds_load_b32 %1, %5\n\t"
                 "ds_load_b32 %2, %6\n\t"
                 "ds_load_b32 %3, %7\n\t"
                 "s_wait_dscnt 0x0"
                 : "=&v"(f0), "=&v"(f1), "=&v"(f2), "=&v"(f3)
                 : "v"(a0), "v"(a1), "v"(a2), "v"(a3)
                 : "memory");
}

__device__ __forceinline__ float lds_gather1(unsigned a0) {
    float f;
    asm volatile("ds_load_b32 %0, %1\n\t"
                 "s_wait_dscnt 0x0"
                 : "=&v"(f)
                 : "v"(a0)
                 : "memory");
    return f;
}

// --------------------------------------------------------- pass 3: apply
// Stage the 256 KB factor table into LDS (CDNA5 WGP has 320 KB) with the
// gfx1250 async-to-LDS path (b128: 512 B per wave-instruction), then gather
// with banked ds_load_b32 instead of hammering L2 with 33.5M random 4B reads.
__global__ __launch_bounds__(1024)
void gl_apply_kernel(const float4* __restrict__ c4,
                     const int4*  __restrict__ g4,
                     const float* __restrict__ fac_g,
                     float4* __restrict__ out4,
                     int n4, int n) {
    // 256 KB, 16B-aligned for B128 async writes; sole LDS object.
    __shared__ __align__(16) float fac[NGROUPS];

    // Wave-relative LDS byte address of the table: low 32 bits of the flat
    // pointer. Taking the address also escapes `fac` to the asm blocks.
    const unsigned fac_base = (unsigned)(uintptr_t)&fac[0];

    // Cooperative async copy: 16384 float4 / 1024 threads = 16 per thread.
    for (int idx4 = threadIdx.x; idx4 < NGROUPS / 4; idx4 += blockDim.x) {
        unsigned     lds_addr = fac_base + (unsigned)idx4 * 16u;
        const float* gp       = fac_g + idx4 * 4;
        asm volatile("global_load_async_to_lds_b128 %0, %1, off"
                     :: "v"(lds_addr), "v"(gp)
                     : "memory");
    }
    // Each wave waits for its own async ops; barrier covers the whole table.
    asm volatile("s_wait_asynccnt 0x0" ::: "memory");
    __syncthreads();

    const int stride = gridDim.x * blockDim.x;
    int tid = blockIdx.x * blockDim.x + threadIdx.x;

    for (int i = tid; i < n4; i += stride) {
        float4 c = c4[i];
        int4   g = g4[i];
        __builtin_prefetch(&c4[i + stride], 0, 3);
        __builtin_prefetch(&g4[i + stride], 0, 3);
        float f0, f1, f2, f3;
        lds_gather4(fac_base + (unsigned)g.x * 4u,
                    fac_base + (unsigned)g.y * 4u,
                    fac_base + (unsigned)g.z * 4u,
                    fac_base + (unsigned)g.w * 4u,
                    f0, f1, f2, f3);
        float4 o;
        o.x = c.x * f0;
        o.y = c.y * f1;
        o.z = c.z * f2;
        o.w = c.w * f3;
        out4[i] = o;
    }
    // scalar tail (N = 2^25 so normally empty)
    const float* c1 = (const float*)c4;
    const int*   g1 = (const int*)g4;
    float*       o1 = (float*)out4;
    for (int i = n4 * 4 + tid; i < n; i += stride) {
        o1[i] = c1[i] * lds_gather1(fac_base + (unsigned)g1[i] * 4u);
    }
}

// ---------------------------------------------------------------- launcher
extern "C" void kernel_launch(void* const* d_in, const int* in_sizes, int n_in,
                              void* d_out, int out_size, void* d_ws, size_t ws_size,
                              hipStream_t stream) {
    const float* coef   = (const float*)d_in[0];
    const int*   groups = (const int*)d_in[1];
    float*       out    = (float*)d_out;
    float*       ws     = (float*)d_ws;      // NGROUPS floats: sumsq -> factor

    int n  = in_sizes[0];
    int n4 = n / 4;

    gl_zero_kernel<<<NGROUPS / 256, 256, 0, stream>>>(ws);
    gl_sumsq_kernel<<<2048, 256, 0, stream>>>(
        (const float4*)coef, (const int4*)groups, ws, n4, n);
    gl_factor_kernel<<<NGROUPS / 256, 256, 0, stream>>>(ws);
    gl_apply_kernel<<<512, 1024, 0, stream>>>(
        (const float4*)coef, (const int4*)groups, ws, (float4*)out, n4, n);
}